// ResidualGATBlock_22136261443949
// MI455X (gfx1250) — compile-verified
//
#include <hip/hip_runtime.h>
#include <hip/hip_bf16.h>

// ResidualGATBlock for MI455X (gfx1250, wave32).
// Pipeline per layer:
//   1) xl = x @ Wl, xr = x @ Wr         -- fp32 WMMA (V_WMMA_F32_16X16X4_F32)
//   2) edge pass A: leaky+att dot -> logits, segment max (encoded-uint atomicMax)
//   3) edge pass B: exp(logit - max), segment sum (atomicAdd)
//   4) edge pass C: alpha * xl[src] scatter-add into agg
//   5) column stats + batchnorm apply (+relu / +residual)
// Bias before BN cancels analytically (constant per column) -> dropped.

#define DD 128
#define HH 4

typedef float v2f __attribute__((ext_vector_type(2)));
typedef float v8f __attribute__((ext_vector_type(8)));

__device__ __forceinline__ unsigned enc_f(float f) {
  unsigned u = __float_as_uint(f);
  return (u & 0x80000000u) ? ~u : (u | 0x80000000u);
}
__device__ __forceinline__ float dec_f(unsigned k) {
  unsigned u = (k & 0x80000000u) ? (k & 0x7FFFFFFFu) : ~k;
  return __uint_as_float(u);
}

// ---------------- GEMM: Y[M x 128] = A[M x 128] @ W[128 x 128] ----------------
// One wave per 16-row tile; computes all 8 column tiles (16x16 each).
// fp32 WMMA 16x16x4: A fragment = 2 VGPRs (lane l: M=l%16, K = (l/16)*2 + {0,1}),
// B fragment mirrored (lane l: N=l%16, K=(l/16)*2+{0,1}), C/D = 8 VGPRs.
__global__ __launch_bounds__(128) void gemm128_wmma(const float* __restrict__ A,
                                                    const float* __restrict__ W,
                                                    float* __restrict__ Y,
                                                    int mtiles, int nrows) {
  int wave = (blockIdx.x << 2) + (threadIdx.x >> 5);
  if (wave >= mtiles) return;
  int l = threadIdx.x & 31;
  int mr = l & 15;
  int hi = l >> 4;      // 0 or 1
  int kh = hi << 1;     // 0 or 2
  int row = wave * 16 + mr;
  int rowc = row < nrows ? row : (nrows - 1);
  const float* arow = A + (size_t)rowc * DD;

  v8f acc[8] = {};
  for (int k = 0; k < DD; k += 4) {
    v2f a;
    a.x = arow[k + kh];
    a.y = arow[k + kh + 1];
    const float* wp0 = W + (size_t)(k + kh) * DD + mr;
#pragma unroll
    for (int nt = 0; nt < 8; ++nt) {
      v2f b;
      b.x = wp0[nt * 16];
      b.y = wp0[nt * 16 + DD];
      acc[nt] = __builtin_amdgcn_wmma_f32_16x16x4_f32(
          false, a, false, b, (short)0, acc[nt], false, false);
    }
  }
  // D layout: VGPR r -> M = 16*wave + hi*8 + r, N = nt*16 + (l%16)
  int mbase = wave * 16 + (hi << 3);
#pragma unroll
  for (int nt = 0; nt < 8; ++nt) {
#pragma unroll
    for (int r = 0; r < 8; ++r) {
      int rr = mbase + r;
      if (rr < nrows) Y[(size_t)rr * DD + nt * 16 + mr] = acc[nt][r];
    }
  }
}

// ---------------- Edge pass A: logits + segment max ----------------
// One wave per edge; lane owns 4 channels (float4). Head h = lane/8.
__global__ __launch_bounds__(256) void edge_logits_kernel(
    const float* __restrict__ xl, const float* __restrict__ xr,
    const int* __restrict__ esrc, const int* __restrict__ edst,
    int Etot, int Eorig, const float* __restrict__ att,
    float* __restrict__ logits, unsigned* __restrict__ mkey) {
  int e = blockIdx.x * 8 + (threadIdx.x >> 5);
  if (e >= Etot) return;
  int l = threadIdx.x & 31;
  int s, d;
  if (e < Eorig) { s = esrc[e]; d = edst[e]; } else { s = e - Eorig; d = s; }
  float4 a = ((const float4*)(xl + (size_t)s * DD))[l];
  float4 b = ((const float4*)(xr + (size_t)d * DD))[l];
  float4 w = ((const float4*)att)[l];
  float e0 = a.x + b.x; e0 = e0 > 0.f ? e0 : 0.2f * e0;
  float e1 = a.y + b.y; e1 = e1 > 0.f ? e1 : 0.2f * e1;
  float e2 = a.z + b.z; e2 = e2 > 0.f ? e2 : 0.2f * e2;
  float e3 = a.w + b.w; e3 = e3 > 0.f ? e3 : 0.2f * e3;
  float sum = e0 * w.x + e1 * w.y + e2 * w.z + e3 * w.w;
  sum += __shfl_xor(sum, 1, 32);
  sum += __shfl_xor(sum, 2, 32);
  sum += __shfl_xor(sum, 4, 32);
  if ((l & 7) == 0) {
    int h = l >> 3;
    logits[(size_t)e * HH + h] = sum;
    atomicMax(&mkey[(size_t)d * HH + h], enc_f(sum));
  }
}

// ---------------- Edge pass B: exp + segment sum ----------------
__global__ __launch_bounds__(256) void edge_exp_kernel(
    float* __restrict__ logits, const unsigned* __restrict__ mkey,
    const int* __restrict__ edst, int Etot, int Eorig,
    float* __restrict__ den) {
  int i = blockIdx.x * 256 + threadIdx.x;
  if (i >= Etot * HH) return;
  int e = i >> 2, h = i & 3;
  int d = (e < Eorig) ? edst[e] : (e - Eorig);
  float m = dec_f(mkey[(size_t)d * HH + h]);
  float ex = __expf(logits[i] - m);
  logits[i] = ex;
  atomicAdd(&den[(size_t)d * HH + h], ex);
}

// ---------------- Edge pass C: aggregate alpha * xl[src] ----------------
__global__ __launch_bounds__(256) void edge_agg_kernel(
    const float* __restrict__ xl, const float* __restrict__ logits,
    const float* __restrict__ den,
    const int* __restrict__ esrc, const int* __restrict__ edst,
    int Etot, int Eorig, float* __restrict__ agg) {
  int e = blockIdx.x * 8 + (threadIdx.x >> 5);
  if (e >= Etot) return;
  int l = threadIdx.x & 31;
  int s, d;
  if (e < Eorig) { s = esrc[e]; d = edst[e]; } else { s = e - Eorig; d = s; }
  int h = l >> 3;
  float alpha = logits[(size_t)e * HH + h] / den[(size_t)d * HH + h];
  float4 v = ((const float4*)(xl + (size_t)s * DD))[l];
  float* o = agg + (size_t)d * DD + l * 4;
  atomicAdd(o + 0, alpha * v.x);
  atomicAdd(o + 1, alpha * v.y);
  atomicAdd(o + 2, alpha * v.z);
  atomicAdd(o + 3, alpha * v.w);
}

// ---------------- Column statistics for batchnorm ----------------
__global__ __launch_bounds__(128) void col_stats_kernel(
    const float* __restrict__ Y, int nrows,
    float* __restrict__ colsum, float* __restrict__ colsumsq) {
  int ch = threadIdx.x;
  int r0 = blockIdx.x * 256;
  int r1 = r0 + 256; if (r1 > nrows) r1 = nrows;
  float s = 0.f, s2 = 0.f;
  for (int r = r0; r < r1; ++r) {
    float v = Y[(size_t)r * DD + ch];
    s += v; s2 += v * v;
  }
  atomicAdd(&colsum[ch], s);
  atomicAdd(&colsumsq[ch], s2);
}

// ---------------- BN apply (+optional residual, always relu) ----------------
__global__ __launch_bounds__(256) void bn_apply_kernel(
    const float* __restrict__ agg, const float* __restrict__ colsum,
    const float* __restrict__ colsumsq, const float* __restrict__ gamma,
    const float* __restrict__ beta, const float* __restrict__ residual,
    float* __restrict__ out, int total, float invN, int use_res) {
  int i = blockIdx.x * 256 + threadIdx.x;
  if (i >= total) return;
  int ch = i & (DD - 1);
  float mu = colsum[ch] * invN;
  float var = colsumsq[ch] * invN - mu * mu;
  float v = gamma[ch] * (agg[i] - mu) * rsqrtf(var + 1e-5f) + beta[ch];
  if (use_res) v += residual[i];
  out[i] = fmaxf(v, 0.0f);
}

extern "C" void kernel_launch(void* const* d_in, const int* in_sizes, int n_in,
                              void* d_out, int out_size, void* d_ws, size_t ws_size,
                              hipStream_t stream) {
  const float* x      = (const float*)d_in[0];
  const int*   eidx   = (const int*)  d_in[1];
  const float* Wl1    = (const float*)d_in[2];
  const float* Wr1    = (const float*)d_in[3];
  const float* att1   = (const float*)d_in[4];
  const float* gamma1 = (const float*)d_in[6];
  const float* beta1  = (const float*)d_in[7];
  const float* Wl2    = (const float*)d_in[8];
  const float* Wr2    = (const float*)d_in[9];
  const float* att2   = (const float*)d_in[10];
  const float* gamma2 = (const float*)d_in[12];
  const float* beta2  = (const float*)d_in[13];

  int N = in_sizes[0] / DD;
  int E = in_sizes[1] / 2;
  int Etot = E + N;                 // self-loops appended
  const int* esrc = eidx;
  const int* edst = eidx + E;

  size_t nd = (size_t)N * DD;
  float*    ws       = (float*)d_ws;
  float*    xl       = ws;
  float*    xr       = xl + nd;
  float*    agg      = xr + nd;
  float*    logits   = agg + nd;
  unsigned* mkey     = (unsigned*)(logits + (size_t)Etot * HH);
  float*    den      = (float*)(mkey + (size_t)N * HH);
  float*    colsum   = den + (size_t)N * HH;
  float*    colsumsq = colsum + DD;
  float*    hbuf     = (float*)d_out;   // layer-1 output staged in d_out

  int mtiles     = (N + 15) / 16;
  int gemmBlocks = (mtiles + 3) / 4;
  int edgeBlocks = (Etot + 7) / 8;
  int ehBlocks   = (Etot * HH + 255) / 256;
  int bnBlocks   = ((int)nd + 255) / 256;
  int csBlocks   = (N + 255) / 256;
  float invN     = 1.0f / (float)N;

  for (int layer = 0; layer < 2; ++layer) {
    const float* xin   = (layer == 0) ? x     : hbuf;
    const float* Wl    = (layer == 0) ? Wl1   : Wl2;
    const float* Wr    = (layer == 0) ? Wr1   : Wr2;
    const float* att   = (layer == 0) ? att1  : att2;
    const float* gamma = (layer == 0) ? gamma1 : gamma2;
    const float* beta  = (layer == 0) ? beta1  : beta2;

    gemm128_wmma<<<gemmBlocks, 128, 0, stream>>>(xin, Wl, xl, mtiles, N);
    gemm128_wmma<<<gemmBlocks, 128, 0, stream>>>(xin, Wr, xr, mtiles, N);

    hipMemsetAsync(agg,  0, nd * sizeof(float), stream);
    hipMemsetAsync(mkey, 0, (size_t)N * HH * sizeof(unsigned), stream);
    hipMemsetAsync(den,  0, (size_t)N * HH * sizeof(float), stream);
    hipMemsetAsync(colsum, 0, 2 * DD * sizeof(float), stream);

    edge_logits_kernel<<<edgeBlocks, 256, 0, stream>>>(xl, xr, esrc, edst, Etot, E,
                                                       att, logits, mkey);
    edge_exp_kernel<<<ehBlocks, 256, 0, stream>>>(logits, mkey, edst, Etot, E, den);
    edge_agg_kernel<<<edgeBlocks, 256, 0, stream>>>(xl, logits, den, esrc, edst,
                                                    Etot, E, agg);
    col_stats_kernel<<<csBlocks, 128, 0, stream>>>(agg, N, colsum, colsumsq);
    bn_apply_kernel<<<bnBlocks, 256, 0, stream>>>(agg, colsum, colsumsq, gamma, beta,
                                                  (layer == 0) ? (const float*)nullptr : x,
                                                  hbuf, (int)nd, invN, layer);
  }
}